// VoronoiValues_81286551044461
// MI455X (gfx1250) — compile-verified
//
#include <hip/hip_runtime.h>
#include <math.h>

// VoronoiValues for MI455X (gfx1250, wave32).
// queries: [N=16384,3] f32, vpoints: [M=6000,3] f32
// out: [N + M*10] f32  = concat(sq_min[N], norm[M*10])
//
// Distance tiles via V_WMMA_F32_16X16X4_F32:
//   A row  = [x, y, z, 1]           (rows = points being ranked)
//   B col  = [-2x, -2y, -2z, |v|^2] (cols = candidate vpoints)
//   D[m][n] = |v_n|^2 - 2 p_m.v_n = d2(m,n) - |p_m|^2  (row-constant shift ->
//   identical per-row ranking; shift re-added where true values are needed)
//
// Vpoint chunks are double-buffered into LDS with the Tensor Data Mover:
// issue TDM for chunk c+1, process chunk c (125 WMMA tiles hide the DMA),
// drain s_wait_tensorcnt at the top of the next iteration.

#define TK      11          // top-(KEDGE+1) / top-(KNN+1)
#define REP     100.0f
#define CHUNK   2000        // vpoints per LDS buffer (multiple of 16)
#define WPB     8           // waves per block
#define BLOCK   (WPB * 32)
#define TSTRIDE 17          // padded row stride of transpose slab (floats)

typedef __attribute__((ext_vector_type(2))) float v2f;
typedef __attribute__((ext_vector_type(8))) float v8f;
typedef __attribute__((ext_vector_type(4))) unsigned int u32x4;
typedef __attribute__((ext_vector_type(8))) int i32x8;
typedef __attribute__((ext_vector_type(4))) int i32x4;

#if __has_include(<hip/amd_detail/amd_gfx1250_TDM.h>)
#define TDM_6ARG 1
#endif

__device__ __forceinline__ void topk_insert(float (&bv)[TK], int (&bi)[TK],
                                            float v, int idx) {
  // bv ascending; stable (equal values keep earlier index, like lax.top_k)
  if (v < bv[TK - 1]) {
    bv[TK - 1] = v;
    bi[TK - 1] = idx;
#pragma unroll
    for (int t = TK - 1; t > 0; --t) {
      if (bv[t] < bv[t - 1]) {
        float tv = bv[t]; bv[t] = bv[t - 1]; bv[t - 1] = tv;
        int   ti = bi[t]; bi[t] = bi[t - 1]; bi[t - 1] = ti;
      }
    }
  }
}

// Issue (without waiting) the staging of raw vpoints [cbase, cbase+cnt) into
// dst, padding to a multiple of 16 points with huge coords.
// TDM path: wave 0 issues tensor_load_to_lds (async, TENSORcnt).
// Fallback: all threads copy synchronously.
__device__ __forceinline__ void tdm_issue(const float* __restrict__ vp,
                                          float* dst, int cbase, int cnt,
                                          int tid, int wave, int lane) {
  const int cntp = (cnt + 15) & ~15;
#if __has_builtin(__builtin_amdgcn_tensor_load_to_lds)
  if (wave == 0) {
    const unsigned long long ga =
        (unsigned long long)(const void*)(vp + (size_t)3 * cbase);
    const unsigned int lds_off = (unsigned int)(size_t)(const void*)dst;
    const unsigned int L = (unsigned int)(cnt * 3);  // dwords, fits 16 bits

    // D# group 0: count=1 | lds_addr | global_addr[56:0] | type=2
    u32x4 g0 = {1u, lds_off, (unsigned int)ga,
                (unsigned int)((ga >> 32) & 0x01FFFFFFull) | 0x80000000u};
    // D# group 1: 1-row 1D tile of L dwords, data_size=4B
    i32x8 g1 = {
        (int)(2u << 16),                                  // data_size=2 (4B)
        (int)((L & 0xFFFFu) << 16),                       // tensor_dim0[15:0]
        (int)(((L >> 16) & 0xFFFFu) | (1u << 16)),        // td0[31:16]|td1=1
        (int)((L & 0xFFFFu) << 16),                       // tile_dim0 = L
        1,                                                // tile_dim1 = 1
        (int)L,                                           // td0_stride[31:0]
        0, 0};
    i32x4 gz = {0, 0, 0, 0};
#ifdef TDM_6ARG
    i32x8 gz8 = {0, 0, 0, 0, 0, 0, 0, 0};
    __builtin_amdgcn_tensor_load_to_lds(g0, g1, gz, gz, gz8, 0);
#else
    __builtin_amdgcn_tensor_load_to_lds(g0, g1, gz, gz, 0);
#endif
    for (int i = cnt * 3 + lane; i < cntp * 3; i += 32)
      dst[i] = 1.0e18f;  // pad points: enormous distance, never selected
  }
#else
  for (int i = tid; i < cnt * 3; i += BLOCK)
    dst[i] = vp[(size_t)3 * cbase + i];
  for (int i = cnt * 3 + tid; i < cntp * 3; i += BLOCK)
    dst[i] = 1.0e18f;
#endif
}

__device__ __forceinline__ void tdm_drain(int wave) {
#if __has_builtin(__builtin_amdgcn_tensor_load_to_lds)
  if (wave == 0) __builtin_amdgcn_s_wait_tensorcnt(0);
#endif
}

// Build B fragment for columns [col16, col16+16) from raw staged points.
__device__ __forceinline__ v2f make_bfrag(const float* buf, int col16,
                                          int half, int mcol) {
  const float* sp = &buf[3 * (col16 + mcol)];
  float x = sp[0], y = sp[1], z = sp[2];
  float v2 = x * x + y * y + z * z;
  float b0 = half ? (-2.0f * z) : (-2.0f * x);  // branchless: cndmask selects
  float b1 = half ? v2 : (-2.0f * y);
  return (v2f){b0, b1};
}

// ---------------------------------------------------------------------------
// Phase 1: per-query top-11 vpoints + edge geometry -> sq_min
// ---------------------------------------------------------------------------
__global__ __launch_bounds__(BLOCK) void voronoi_query_kernel(
    const float* __restrict__ q, const float* __restrict__ vp,
    float* __restrict__ out, int N, int M) {
  __shared__ float svraw[2][3 * CHUNK];              // double-buffered points
  __shared__ float tdist[WPB][32 * TSTRIDE];         // per-wave transpose slab

  const int tid  = threadIdx.x;
  const int wave = tid >> 5;
  const int lane = tid & 31;
  const int half = lane >> 4;          // 0 -> K={0,1}, 1 -> K={2,3}
  const int mcol = lane & 15;
  const int rbase = blockIdx.x * BLOCK + wave * 32;  // 32 query rows per wave

  v2f a0, a1;
  {
    int m0 = min(rbase + mcol, N - 1);
    int m1 = min(rbase + mcol + 16, N - 1);
    float x0 = q[3 * m0], y0 = q[3 * m0 + 1], z0 = q[3 * m0 + 2];
    float x1 = q[3 * m1], y1 = q[3 * m1 + 1], z1 = q[3 * m1 + 2];
    a0 = half ? (v2f){z0, 1.0f} : (v2f){x0, y0};
    a1 = half ? (v2f){z1, 1.0f} : (v2f){x1, y1};
  }

  float bv[TK]; int bi[TK];
#pragma unroll
  for (int t = 0; t < TK; ++t) { bv[t] = 3.0e38f; bi[t] = 0; }

  float* tslab = &tdist[wave][0];

  tdm_issue(vp, svraw[0], 0, min(CHUNK, M), tid, wave, lane);
  int pc = 0;
  for (int cbase = 0; cbase < M; cbase += CHUNK) {
    const int cnt  = min(CHUNK, M - cbase);
    const int cntp = (cnt + 15) & ~15;
    tdm_drain(wave);
    __syncthreads();                   // buf[pc] visible to all waves
    const int nb = cbase + CHUNK;
    if (nb < M)                        // overlap next DMA with this compute
      tdm_issue(vp, svraw[pc ^ 1], nb, min(CHUNK, M - nb), tid, wave, lane);
    const float* buf = svraw[pc];

    const int ntile = cntp >> 4;
    for (int jt = 0; jt < ntile; ++jt) {
      v2f b = make_bfrag(buf, jt * 16, half, mcol);

      v8f c0 = {}, c1 = {};
      c0 = __builtin_amdgcn_wmma_f32_16x16x4_f32(false, a0, false, b,
                                                 (short)0, c0, false, false);
      c1 = __builtin_amdgcn_wmma_f32_16x16x4_f32(false, a1, false, b,
                                                 (short)0, c1, false, false);

      // transpose D tiles through LDS: VGPR r -> row r + 8*half (+16 for c1)
#pragma unroll
      for (int r = 0; r < 8; ++r) {
        int row = r + (half << 3);
        tslab[row * TSTRIDE + mcol]        = c0[r];
        tslab[(row + 16) * TSTRIDE + mcol] = c1[r];
      }
      asm volatile("s_wait_dscnt 0x0" ::: "memory");  // wave-local DS RAW

      const int gcol = cbase + jt * 16;
#pragma unroll
      for (int k = 0; k < 16; ++k) {
        float d = tslab[lane * TSTRIDE + k];
        topk_insert(bv, bi, d, gcol + k);
      }
      asm volatile("" ::: "memory");  // keep next tile's stores after reads
    }
    pc ^= 1;
    __syncthreads();                   // all readers done before overwrite
  }

  // geometry: sq_min over 10 edges from nearest vpoint
  const int qrow = rbase + lane;
  const int qc   = min(qrow, N - 1);
  float qx = q[3 * qc], qy = q[3 * qc + 1], qz = q[3 * qc + 2];
  int   i0 = bi[0];
  float px = vp[3 * i0], py = vp[3 * i0 + 1], pz = vp[3 * i0 + 2];
  float cx = qx - px, cy = qy - py, cz = qz - pz;
  float best = 3.0e38f;
#pragma unroll
  for (int e = 1; e < TK; ++e) {
    int ie = bi[e];
    float ex = vp[3 * ie] - px, ey = vp[3 * ie + 1] - py, ez = vp[3 * ie + 2] - pz;
    float el = sqrtf(ex * ex + ey * ey + ez * ez);
    float vl = (cx * ex + cy * ey + cz * ez) / el;
    float s  = vl - 0.5f * el;
    best = fminf(best, s * s);
  }
  if (qrow < N) out[qrow] = best;
}

// ---------------------------------------------------------------------------
// Phase 2: per-vpoint 10 smallest non-self distances (ascending) -> repulsion
// ---------------------------------------------------------------------------
__global__ __launch_bounds__(BLOCK) void voronoi_repulsion_kernel(
    const float* __restrict__ vp, float* __restrict__ out, int N, int M) {
  __shared__ float svraw[2][3 * CHUNK];
  __shared__ float tdist[WPB][32 * TSTRIDE];

  const int tid  = threadIdx.x;
  const int wave = tid >> 5;
  const int lane = tid & 31;
  const int half = lane >> 4;
  const int mcol = lane & 15;
  const int rbase = blockIdx.x * BLOCK + wave * 32;

  v2f a0, a1;
  {
    int m0 = min(rbase + mcol, M - 1);
    int m1 = min(rbase + mcol + 16, M - 1);
    float x0 = vp[3 * m0], y0 = vp[3 * m0 + 1], z0 = vp[3 * m0 + 2];
    float x1 = vp[3 * m1], y1 = vp[3 * m1 + 1], z1 = vp[3 * m1 + 2];
    a0 = half ? (v2f){z0, 1.0f} : (v2f){x0, y0};
    a1 = half ? (v2f){z1, 1.0f} : (v2f){x1, y1};
  }

  float bv[TK]; int bi[TK];
#pragma unroll
  for (int t = 0; t < TK; ++t) { bv[t] = 3.0e38f; bi[t] = 0; }

  float* tslab = &tdist[wave][0];

  tdm_issue(vp, svraw[0], 0, min(CHUNK, M), tid, wave, lane);
  int pc = 0;
  for (int cbase = 0; cbase < M; cbase += CHUNK) {
    const int cnt  = min(CHUNK, M - cbase);
    const int cntp = (cnt + 15) & ~15;
    tdm_drain(wave);
    __syncthreads();
    const int nb = cbase + CHUNK;
    if (nb < M)
      tdm_issue(vp, svraw[pc ^ 1], nb, min(CHUNK, M - nb), tid, wave, lane);
    const float* buf = svraw[pc];

    const int ntile = cntp >> 4;
    for (int jt = 0; jt < ntile; ++jt) {
      v2f b = make_bfrag(buf, jt * 16, half, mcol);

      v8f c0 = {}, c1 = {};
      c0 = __builtin_amdgcn_wmma_f32_16x16x4_f32(false, a0, false, b,
                                                 (short)0, c0, false, false);
      c1 = __builtin_amdgcn_wmma_f32_16x16x4_f32(false, a1, false, b,
                                                 (short)0, c1, false, false);

#pragma unroll
      for (int r = 0; r < 8; ++r) {
        int row = r + (half << 3);
        tslab[row * TSTRIDE + mcol]        = c0[r];
        tslab[(row + 16) * TSTRIDE + mcol] = c1[r];
      }
      asm volatile("s_wait_dscnt 0x0" ::: "memory");

      const int gcol = cbase + jt * 16;
#pragma unroll
      for (int k = 0; k < 16; ++k) {
        float d = tslab[lane * TSTRIDE + k];
        topk_insert(bv, bi, d, gcol + k);  // shifted by -|v_row|^2 (rank-safe)
      }
      asm volatile("" ::: "memory");
    }
    pc ^= 1;
    __syncthreads();
  }

  const int row = rbase + lane;
  if (row < M) {
    float rx = vp[3 * row], ry = vp[3 * row + 1], rz = vp[3 * row + 2];
    float r2 = rx * rx + ry * ry + rz * rz;   // undo the row-constant shift
    // bv[0] ~ self (d2 ~ 0); ranks 1..10 ascending, matching top_k order
#pragma unroll
    for (int e = 1; e < TK; ++e) {
      float d2 = bv[e] + r2;
      out[N + row * (TK - 1) + (e - 1)] = __expf(-REP * d2) * (1.0f / REP);
    }
  }
}

extern "C" void kernel_launch(void* const* d_in, const int* in_sizes, int n_in,
                              void* d_out, int out_size, void* d_ws, size_t ws_size,
                              hipStream_t stream) {
  const float* q  = (const float*)d_in[0];
  const float* vp = (const float*)d_in[1];
  float* out = (float*)d_out;
  const int N = in_sizes[0] / 3;   // 16384
  const int M = in_sizes[1] / 3;   // 6000

  const int gq = (N + BLOCK - 1) / BLOCK;  // 256 query rows per block
  const int gv = (M + BLOCK - 1) / BLOCK;  // 256 vpoint rows per block
  voronoi_query_kernel<<<gq, BLOCK, 0, stream>>>(q, vp, out, N, M);
  voronoi_repulsion_kernel<<<gv, BLOCK, 0, stream>>>(vp, out, N, M);
}